// MHCMapping_41223096107387
// MI455X (gfx1250) — compile-verified
//
#include <hip/hip_runtime.h>
#include <math.h>

typedef __attribute__((ext_vector_type(2))) float v2f;
typedef __attribute__((ext_vector_type(4))) float v4f;
typedef __attribute__((ext_vector_type(8))) float v8f;

#define DFLAT 4096
#define NROWS 16384      // B*T = 4*4096
#define NCOLS 24         // 16 res + 4 pre + 4 post
#define SPLITS 4
#define KSPLIT (DFLAT / SPLITS)   // 1024
#define RMS_EPS 1e-6f
#define SK_EPS  1e-6f

// ---------------------------------------------------------------------------
// Kernel 0: Wcat[32][4096] = concat(W_res[16], W_pre[4], W_post[4], zeros[8]) * g
// ---------------------------------------------------------------------------
__global__ void prep_weights(const float* __restrict__ g,
                             const float* __restrict__ W_pre,
                             const float* __restrict__ W_post,
                             const float* __restrict__ W_res,
                             float* __restrict__ Wcat) {
  int idx = blockIdx.x * blockDim.x + threadIdx.x;   // 0 .. 32*4096-1
  int m = idx >> 12;
  int k = idx & (DFLAT - 1);
  float w;
  if      (m < 16) w = W_res [(m     ) * DFLAT + k];
  else if (m < 20) w = W_pre [(m - 16) * DFLAT + k];
  else if (m < 24) w = W_post[(m - 20) * DFLAT + k];
  else             w = 0.0f;
  Wcat[idx] = w * g[k];
}

// ---------------------------------------------------------------------------
// Kernel 1: 4 waves per 16-row tile (split-K), 2 tiles per 256-thread block.
// Each wave: v_wmma_f32_16x16x4_f32 over its 1024-wide K range, plus per-lane
// sum-of-squares. Partials combined through LDS; split-0 wave applies
// rsqrt(mean(x^2)+eps) and writes the 24 normalized dots per row.
//
// K grouped by 8: low lanes hold K {k..k+3}, high lanes {k+4..k+7}; WMMA #0
// consumes (.x,.y), WMMA #1 consumes (.z,.w). Identical K permutation on A
// and B keeps the dot products exact.
// ---------------------------------------------------------------------------
__global__ void __launch_bounds__(256) gemv_wmma(const float* __restrict__ X,
                                                 const float* __restrict__ Wcat,
                                                 float* __restrict__ dots) {
  const int lane  = threadIdx.x & 31;
  const int wave  = threadIdx.x >> 5;       // 0..7
  const int tib   = wave >> 2;              // tile in block: 0..1
  const int split = wave & 3;               // K split: 0..3
  const int tile  = blockIdx.x * 2 + tib;   // 1024 tiles of 16 rows
  const int rowBase = tile * 16;
  const int hi = lane >> 4;                 // K half select
  const int m  = lane & 15;                 // A: row in tile, B: column
  const int kBase = split * KSPLIT;

  const float* aPtr  = X    + (size_t)(rowBase + m) * DFLAT + kBase + (hi ? 4 : 0);
  const float* b0Ptr = Wcat + (size_t)(m     ) * DFLAT + kBase + (hi ? 4 : 0);
  const float* b1Ptr = Wcat + (size_t)(16 + m) * DFLAT + kBase + (hi ? 4 : 0);

  v8f c0 = {};      // 16 W_res columns
  v8f c1 = {};      // 4 W_pre + 4 W_post + 8 zero columns
  float ss = 0.0f;  // per-lane partial sum of squares (raw x)

#pragma unroll 4
  for (int k = 0; k < KSPLIT; k += 8) {
    __builtin_prefetch(aPtr + k + 1024, 0, 1);             // stream X ahead
    v4f av  = __builtin_nontemporal_load((const v4f*)(aPtr  + k));
    v4f b0v = *(const v4f*)(b0Ptr + k);
    v4f b1v = *(const v4f*)(b1Ptr + k);

    ss += av.x * av.x + av.y * av.y + av.z * av.z + av.w * av.w;

    v2f a01 = {av.x,  av.y},  a23 = {av.z,  av.w};
    v2f b0a = {b0v.x, b0v.y}, b0b = {b0v.z, b0v.w};
    v2f b1a = {b1v.x, b1v.y}, b1b = {b1v.z, b1v.w};

    c0 = __builtin_amdgcn_wmma_f32_16x16x4_f32(false, a01, false, b0a, (short)0, c0, false, false);
    c1 = __builtin_amdgcn_wmma_f32_16x16x4_f32(false, a01, false, b1a, (short)0, c1, false, false);
    c0 = __builtin_amdgcn_wmma_f32_16x16x4_f32(false, a23, false, b0b, (short)0, c0, false, false);
    c1 = __builtin_amdgcn_wmma_f32_16x16x4_f32(false, a23, false, b1b, (short)0, c1, false, false);
  }

  // ---- combine split-K partials through LDS ----
  __shared__ float part[2][SPLITS][16][32];   // 16 KB
  __shared__ float pss [2][SPLITS][32];       // 1 KB
#pragma unroll
  for (int r = 0; r < 8; ++r) {
    part[tib][split][r    ][lane] = c0[r];
    part[tib][split][8 + r][lane] = c1[r];
  }
  pss[tib][split][lane] = ss;
  __syncthreads();

  if (split == 0) {
    float ssT = pss[tib][0][lane] + pss[tib][1][lane]
              + pss[tib][2][lane] + pss[tib][3][lane];
    ssT += __shfl_xor(ssT, 16, 32);           // add the other K half of this row
    float rms = rsqrtf(ssT * (1.0f / DFLAT) + RMS_EPS);  // valid for row lane&15

    // C layout: VGPR r -> (row r, col m) low lanes / (row r+8, col m) high lanes
#pragma unroll
    for (int r = 0; r < 8; ++r) {
      float s0 = part[tib][0][r][lane] + part[tib][1][r][lane]
               + part[tib][2][r][lane] + part[tib][3][r][lane];
      float s1 = part[tib][0][8 + r][lane] + part[tib][1][8 + r][lane]
               + part[tib][2][8 + r][lane] + part[tib][3][8 + r][lane];
      int rloc = r + (hi ? 8 : 0);
      float rmsr = __shfl(rms, rloc, 32);     // rms of row rloc lives in lane rloc
      int row = rowBase + rloc;
      dots[row * NCOLS + m] = s0 * rmsr;
      if (m < 8) dots[row * NCOLS + 16 + m] = s1 * rmsr;
    }
  }
}

// ---------------------------------------------------------------------------
// Kernel 2: per-row epilogue — sigmoids + 20-iteration Sinkhorn on 4x4.
// Output layout (concatenated flat, reference return order):
//   H_pre  [NROWS*4] | H_post [NROWS*4] | H_res [NROWS*16]
// ---------------------------------------------------------------------------
__global__ void postproc(const float* __restrict__ dots,
                         const float* __restrict__ b_pre,
                         const float* __restrict__ b_post,
                         const float* __restrict__ b_res,
                         const float* __restrict__ alpha_pre,
                         const float* __restrict__ alpha_post,
                         const float* __restrict__ alpha_res,
                         float* __restrict__ out) {
  int row = blockIdx.x * blockDim.x + threadIdx.x;
  if (row >= NROWS) return;

  const float ap = alpha_pre[0], apo = alpha_post[0], ar = alpha_res[0];
  const float* d = dots + (size_t)row * NCOLS;

  float* out_pre  = out;
  float* out_post = out + NROWS * 4;
  float* out_res  = out + NROWS * 8;

#pragma unroll
  for (int n = 0; n < 4; ++n) {
    float hp = ap  * d[16 + n] + b_pre[n];
    out_pre [row * 4 + n] = 1.0f / (1.0f + expf(-hp));
    float ho = apo * d[20 + n] + b_post[n];
    out_post[row * 4 + n] = 2.0f / (1.0f + expf(-ho));
  }

  float P[16];
#pragma unroll
  for (int e = 0; e < 16; ++e) {
    float h = ar * d[e] + b_res[e];
    h = fminf(fmaxf(h, -15.0f), 15.0f);
    P[e] = expf(h);
  }
  for (int it = 0; it < 20; ++it) {
#pragma unroll
    for (int j = 0; j < 4; ++j) {          // column normalize (axis -2)
      float inv = 1.0f / (P[j] + P[4 + j] + P[8 + j] + P[12 + j] + SK_EPS);
      P[j] *= inv; P[4 + j] *= inv; P[8 + j] *= inv; P[12 + j] *= inv;
    }
#pragma unroll
    for (int i = 0; i < 4; ++i) {          // row normalize (axis -1)
      float inv = 1.0f / (P[4 * i] + P[4 * i + 1] + P[4 * i + 2] + P[4 * i + 3] + SK_EPS);
      P[4 * i] *= inv; P[4 * i + 1] *= inv; P[4 * i + 2] *= inv; P[4 * i + 3] *= inv;
    }
  }
#pragma unroll
  for (int e = 0; e < 16; ++e) out_res[row * 16 + e] = P[e];
}

// ---------------------------------------------------------------------------
extern "C" void kernel_launch(void* const* d_in, const int* in_sizes, int n_in,
                              void* d_out, int out_size, void* d_ws, size_t ws_size,
                              hipStream_t stream) {
  const float* X          = (const float*)d_in[0];
  const float* g          = (const float*)d_in[1];
  const float* W_pre      = (const float*)d_in[2];
  const float* W_post     = (const float*)d_in[3];
  const float* W_res      = (const float*)d_in[4];
  const float* b_pre      = (const float*)d_in[5];
  const float* b_post     = (const float*)d_in[6];
  const float* b_res      = (const float*)d_in[7];
  const float* alpha_pre  = (const float*)d_in[8];
  const float* alpha_post = (const float*)d_in[9];
  const float* alpha_res  = (const float*)d_in[10];
  float* out  = (float*)d_out;

  float* Wcat = (float*)d_ws;                 // 32*4096 floats = 512 KB
  float* dots = Wcat + 32 * DFLAT;            // 16384*24 floats = 1.5 MB

  prep_weights<<<(32 * DFLAT) / 256, 256, 0, stream>>>(g, W_pre, W_post, W_res, Wcat);
  gemv_wmma<<<(NROWS / 16) / 2, 256, 0, stream>>>(X, Wcat, dots);   // 512 blocks, 8 waves
  postproc<<<NROWS / 256, 256, 0, stream>>>(dots, b_pre, b_post, b_res,
                                            alpha_pre, alpha_post, alpha_res, out);
}